// TransformerEmbedder_28802050687526
// MI455X (gfx1250) — compile-verified
//
#include <hip/hip_runtime.h>

// Segment-mean as a one-hot WMMA matmul on gfx1250 (wave32).
// out[b,t,:] = (1/count[t]) * sum_{s in segment t} hidden[b,s,:]
// Sorted segment_ids => contiguous subtoken row ranges per token (binary
// search, no atomics). A-tile built in registers from the search results,
// B-tile streamed from HBM with 32-bit offsets off an SGPR base, K-loop
// unrolled x2 over two independent accumulators to pipeline the XDL.

typedef __attribute__((ext_vector_type(2))) float v2f;
typedef __attribute__((ext_vector_type(8))) float v8f;

static constexpr int kB = 64;
static constexpr int kS = 512;
static constexpr int kD = 768;
static constexpr int kT = 384;
static constexpr int kTT = kT / 16;  // 24 token tiles
static constexpr int kDT = kD / 16;  // 48 dim tiles
static constexpr int kWavesPerBlock = 8;

__device__ __forceinline__ int lower_bound_i32(const int* __restrict__ a, int n, int key) {
    int lo = 0, len = n;
    while (len > 0) {
        int h   = len >> 1;
        int mid = lo + h;
        if (a[mid] < key) { lo = mid + 1; len -= h + 1; }
        else              { len = h; }
    }
    return lo;
}

// One 16x16x4 K-chunk: rows s..s+3. A entry = inv iff row belongs to this
// lane's token; clamped tail rows have A==0 so their loads contribute nothing.
__device__ __forceinline__ v8f chunk_wmma(int s, int half, int lb0, int lb1, float inv,
                                          const float* __restrict__ hb, unsigned col,
                                          v8f c) {
    const int r0 = s + 2 * half;      // K = 2*half   row
    const int r1 = r0 + 1;            // K = 2*half+1 row
    v2f a;
    a.x = (r0 >= lb0 && r0 < lb1) ? inv : 0.0f;
    a.y = (r1 >= lb0 && r1 < lb1) ? inv : 0.0f;
    const unsigned c0 = (unsigned)((r0 < kS - 1) ? r0 : (kS - 1));
    const unsigned c1 = (unsigned)((r1 < kS - 1) ? r1 : (kS - 1));
    v2f bm;
    bm.x = hb[c0 * (unsigned)kD + col];   // 32-bit voffset, SGPR base
    bm.y = hb[c1 * (unsigned)kD + col];
    return __builtin_amdgcn_wmma_f32_16x16x4_f32(
        /*neg_a=*/false, a, /*neg_b=*/false, bm,
        /*c_mod=*/(short)0, c, /*reuse_a=*/false, /*reuse_b=*/false);
}

__global__ __launch_bounds__(kWavesPerBlock * 32)
void seg_mean_wmma_kernel(const float* __restrict__ hs,
                          const int*   __restrict__ seg,
                          float*       __restrict__ out) {
    const int wave = blockIdx.x * kWavesPerBlock + (threadIdx.x >> 5);
    const int lane = threadIdx.x & 31;
    if (wave >= kB * kTT * kDT) return;  // wave-uniform; EXEC stays all-1s

    const int dd = wave % kDT;
    const int tt = (wave / kDT) % kTT;
    const int b  = wave / (kDT * kTT);
    const int t0 = tt * 16;
    const int d0 = dd * 16;

    const int m    = lane & 15;  // token row within tile (A: M, C/D & B: N)
    const int half = lane >> 4;  // K striping: vgpr0 -> K=2*half, vgpr1 -> K=2*half+1

    // Per-lane segment range for token t0+m (9-step binary searches, sorted ids).
    const int* srow = seg + b * kS;
    const int lb0 = lower_bound_i32(srow, kS, t0 + m);
    const int lb1 = lower_bound_i32(srow, kS, t0 + m + 1);
    const float inv = (lb1 > lb0) ? (1.0f / (float)(lb1 - lb0)) : 0.0f;

    // Wave-uniform subtoken range covered by this 16-token tile.
    const int s_begin = __builtin_amdgcn_readlane(lb0, 0);
    const int s_end   = __builtin_amdgcn_readlane(lb1, 15);

    // Uniform SGPR base; per-lane 32-bit column offset (buffer < 2^32 bytes).
    const float*   hb  = hs + (size_t)b * kS * kD;
    const unsigned col = (unsigned)(d0 + m);

    // Two independent accumulator chains to pipeline back-to-back WMMAs.
    v8f acc0 = {}, acc1 = {};
    int s = s_begin;
    for (; s + 4 < s_end; s += 8) {
        acc0 = chunk_wmma(s,     half, lb0, lb1, inv, hb, col, acc0);
        acc1 = chunk_wmma(s + 4, half, lb0, lb1, inv, hb, col, acc1);
    }
    if (s < s_end) {  // uniform remainder chunk
        acc0 = chunk_wmma(s, half, lb0, lb1, inv, hb, col, acc0);
    }

    // C/D layout: VGPR r -> token t0 + r + 8*half, column N = m.
    // 32-bit store offsets; r*kD folds into the instruction offset field.
    const unsigned obase =
        ((unsigned)(b * kT + t0 + 8 * half)) * (unsigned)kD + (unsigned)(d0 + m);
#pragma unroll
    for (int r = 0; r < 8; ++r) {
        out[obase + (unsigned)(r * kD)] = acc0[r] + acc1[r];
    }
}

extern "C" void kernel_launch(void* const* d_in, const int* in_sizes, int n_in,
                              void* d_out, int out_size, void* d_ws, size_t ws_size,
                              hipStream_t stream) {
    (void)in_sizes; (void)n_in; (void)out_size; (void)d_ws; (void)ws_size;
    const float* hs  = (const float*)d_in[0];
    const int*   seg = (const int*)d_in[1];
    float*       out = (float*)d_out;

    const int total_waves = kB * kTT * kDT;                                  // 73728
    const int blocks = (total_waves + kWavesPerBlock - 1) / kWavesPerBlock;  // 9216
    seg_mean_wmma_kernel<<<blocks, kWavesPerBlock * 32, 0, stream>>>(hs, seg, out);
}